// GlobalFilter_3384434229879
// MI455X (gfx1250) — compile-verified
//
#include <hip/hip_runtime.h>

typedef __attribute__((ext_vector_type(2))) float v2f;
typedef __attribute__((ext_vector_type(8))) float v8f;

#define P      16
#define CDIM   768
#define NSP    256      // 16*16 spatial points
#define CTILE  16       // channels per block
#define XPITCH 260      // padded pitch for the x/y channel rows in LDS
#define MPITCH 17       // padded pitch for 16x16 matrices in LDS

// One 16x16x16 matmul-accumulate as 4 chained V_WMMA_F32_16X16X4_F32.
// A is read row-major with pitch `ap` (A-operand layout: lane -> row=lane&15,
// k = 2*(lane>>4)+v within each K=4 slice). B is read either non-transposed
// (B[k][n] = B[k*bp+n]) or transposed (B[k][n] = B[n*bp+k]). NEGA folds a
// minus sign into the A slice (f32 WMMA has no A-negate modifier).
template<bool NEGA, bool BT>
__device__ __forceinline__ v8f mac16(v8f acc, const float* __restrict__ A, int ap,
                                     const float* __restrict__ B, int bp, int lane) {
  const int ln = lane & 15;
  const int kb = (lane >> 4) << 1;
#pragma unroll
  for (int s = 0; s < 4; ++s) {
    const int k0 = 4 * s + kb;
    v2f a, b;
    a.x = A[ln * ap + k0];
    a.y = A[ln * ap + k0 + 1];
    if (NEGA) { a.x = -a.x; a.y = -a.y; }
    if (BT) {
      b.x = B[ln * bp + k0];
      b.y = B[ln * bp + k0 + 1];
    } else {
      b.x = B[(k0    ) * bp + ln];
      b.y = B[(k0 + 1) * bp + ln];
    }
    acc = __builtin_amdgcn_wmma_f32_16x16x4_f32(false, a, false, b, (short)0, acc,
                                                false, false);
  }
  return acc;
}

// Store a 16x16 f32 accumulator (C/D layout) to an LDS matrix with pitch MPITCH.
__device__ __forceinline__ void stC(float* __restrict__ buf, v8f acc, int lane) {
  const int ln = lane & 15;
  const int rb = (lane >> 4) << 3;
#pragma unroll
  for (int v = 0; v < 8; ++v) buf[(rb + v) * MPITCH + ln] = acc[v];
}

__global__ __launch_bounds__(256) void global_filter_kernel(
    const float* __restrict__ x, const float* __restrict__ w, float* __restrict__ y) {
  __shared__ float sCr[P * MPITCH];           // Re(F_ortho) = cos(2pi kn/16)/4
  __shared__ float sCi[P * MPITCH];           // Im(F_ortho) = -sin(2pi kn/16)/4
  __shared__ float sX[CTILE * XPITCH];        // x tile, later reused for y tile
  __shared__ float sWk[8][4 * P * MPITCH];    // per-wave scratch: 4 matrices

  const int tid  = threadIdx.x;
  const int lane = tid & 31;
  const int wv   = tid >> 5;
  const int b    = blockIdx.y;
  const int c0   = blockIdx.x * CTILE;

  // Build the ortho DFT matrix (all 256 threads, one element each).
  {
    const int k = tid >> 4, n = tid & 15;
    float s, c;
    __sincosf((float)(k * n) * 0.39269908169872414f /* pi/8 */, &s, &c);
    sCr[k * MPITCH + n] = 0.25f * c;
    sCi[k * MPITCH + n] = -0.25f * s;
  }

  // Coalesced load of x[b, 0:256, c0:c0+16] -> sX[channel][spatial].
  const float* xg = x + (size_t)b * NSP * CDIM + c0;
#pragma unroll
  for (int i = 0; i < 4; ++i) {
    const int idx = tid + i * 256;
    const int n = idx & 255, g = idx >> 8;
    const float4 v = *(const float4*)(xg + (size_t)n * CDIM + 4 * g);
    sX[(4 * g + 0) * XPITCH + n] = v.x;
    sX[(4 * g + 1) * XPITCH + n] = v.y;
    sX[(4 * g + 2) * XPITCH + n] = v.z;
    sX[(4 * g + 3) * XPITCH + n] = v.w;
  }
  __syncthreads();

  float* bufA = &sWk[wv][0];
  float* bufB = &sWk[wv][1 * P * MPITCH];
  float* bufC = &sWk[wv][2 * P * MPITCH];
  float* bufD = &sWk[wv][3 * P * MPITCH];
  const v8f zero = {};

  // Hermitian-extended filter index map: loop-invariant per lane, hoisted out
  // of the channel loop. 32-bit offsets so the backend can use GVS addressing
  // (SGPR base + 32-bit VGPR offset) for the weight fetches.
  int   wbase[8];
  float wsgn[8];
  {
    const int ln = lane & 15;
    const int rb = (lane >> 4) << 3;
#pragma unroll
    for (int v = 0; v < 8; ++v) {
      const int k1 = rb + v, k2 = ln;
      int kk1, kk2;
      if (k2 <= 8) { kk1 = k1;            kk2 = k2;     wsgn[v] =  1.0f; }
      else         { kk1 = (P - k1) & 15; kk2 = P - k2; wsgn[v] = -1.0f; }
      wbase[v] = (kk1 * 9 + kk2) * CDIM * 2;
    }
  }

#pragma unroll 1
  for (int cc = 0; cc < 2; ++cc) {        // 8 waves x 2 channels = 16 channels
    const int cl = wv * 2 + cc;
    const int c2 = (c0 + cl) * 2;
    const float* xt = &sX[cl * XPITCH];   // 16x16 real tile, pitch 16

    // Stage 1: A = F * x   (Ar = Cr x, Ai = Ci x)
    v8f ar = mac16<false, false>(zero, sCr, MPITCH, xt, P, lane);
    v8f ai = mac16<false, false>(zero, sCi, MPITCH, xt, P, lane);
    stC(bufA, ar, lane);
    stC(bufB, ai, lane);

    // Stage 2: X = A * F^T  (Xr = Ar Cr^T - Ai Ci^T, Xi = Ar Ci^T + Ai Cr^T)
    v8f xr = mac16<false, true>(zero, bufA, MPITCH, sCr, MPITCH, lane);
    xr     = mac16<true,  true>(xr,   bufB, MPITCH, sCi, MPITCH, lane);
    v8f xi = mac16<false, true>(zero, bufA, MPITCH, sCi, MPITCH, lane);
    xi     = mac16<false, true>(xi,   bufB, MPITCH, sCr, MPITCH, lane);

    // Stage 3: pointwise multiply with Hermitian-extended complex filter.
    v8f yr, yi;
#pragma unroll
    for (int v = 0; v < 8; ++v) {
      const float2 wv2 = *(const float2*)(w + wbase[v] + c2);
      const float wr = wv2.x;
      const float wi = wsgn[v] * wv2.y;
      yr[v] = wr * xr[v] - wi * xi[v];
      yi[v] = wr * xi[v] + wi * xr[v];
    }
    stC(bufC, yr, lane);
    stC(bufD, yi, lane);

    // Stage 4: B = conj(F) * Y  (Br = Cr Yr + Ci Yi, Bi = Cr Yi - Ci Yr)
    v8f br = mac16<false, false>(zero, sCr, MPITCH, bufC, MPITCH, lane);
    br     = mac16<false, false>(br,   sCi, MPITCH, bufD, MPITCH, lane);
    v8f bi = mac16<false, false>(zero, sCr, MPITCH, bufD, MPITCH, lane);
    bi     = mac16<true,  false>(bi,   sCi, MPITCH, bufC, MPITCH, lane);
    stC(bufA, br, lane);
    stC(bufB, bi, lane);

    // Stage 5: y = Re(B * conj(F)^T) = Br Cr^T + Bi Ci^T
    v8f yy = mac16<false, true>(zero, bufA, MPITCH, sCr, MPITCH, lane);
    yy     = mac16<false, true>(yy,   bufB, MPITCH, sCi, MPITCH, lane);

    // Write result back into this channel's sX row (spatial layout, pitch 16).
    {
      const int ln = lane & 15;
      const int rb = (lane >> 4) << 3;
#pragma unroll
      for (int v = 0; v < 8; ++v)
        sX[cl * XPITCH + (rb + v) * P + ln] = yy[v];
    }
  }
  __syncthreads();

  // Coalesced float4 store of the y tile.
  float* yg = y + (size_t)b * NSP * CDIM + c0;
#pragma unroll
  for (int i = 0; i < 4; ++i) {
    const int idx = tid + i * 256;
    const int n = idx & 255, g = idx >> 8;
    float4 v;
    v.x = sX[(4 * g + 0) * XPITCH + n];
    v.y = sX[(4 * g + 1) * XPITCH + n];
    v.z = sX[(4 * g + 2) * XPITCH + n];
    v.w = sX[(4 * g + 3) * XPITCH + n];
    *(float4*)(yg + (size_t)n * CDIM + 4 * g) = v;
  }
}

extern "C" void kernel_launch(void* const* d_in, const int* in_sizes, int n_in,
                              void* d_out, int out_size, void* d_ws, size_t ws_size,
                              hipStream_t stream) {
  const float* x = (const float*)d_in[0];
  const float* w = (const float*)d_in[1];
  float* y = (float*)d_out;
  const int B = in_sizes[0] / (NSP * CDIM);   // 512
  dim3 grid(CDIM / CTILE, B);                 // 48 x 512 blocks
  global_filter_kernel<<<grid, dim3(256), 0, stream>>>(x, w, y);
}